// SpatialEmbLoss_3d_75118978007385
// MI455X (gfx1250) — compile-verified
//
#include <hip/hip_runtime.h>
#include <hip/hip_bf16.h>
#include <math.h>

#define BATCH 2
#define CH    7
#define DD    48
#define HH    160
#define WW    160
#define VOL   (DD*HH*WW)          /* 1,228,800 */
#define NID   15
#define NPAIR (BATCH*NID)         /* 30 */
#define TPB   256
#define IPT   16
#define TILE  (TPB*IPT)           /* 4096 */
#define NTILES (VOL/TILE)         /* 300 exactly */

static_assert(NTILES * TILE == VOL, "volume must tile exactly");

typedef float v2f __attribute__((ext_vector_type(2)));
typedef float v8f __attribute__((ext_vector_type(8)));

// ---------------------------------------------------------------------------
// Wave-level 64-value f32 sum using V_WMMA_F32_16X16X4_F32.
// Each of the 32 lanes contributes 2 values -> A is the full 16x4 tile.
// B = ones(4x16)  =>  D[M][N] = rowsum(A)[M].  Lanes 0-15 hold rows 0..7 in
// their 8 accumulator VGPRs, lanes 16-31 hold rows 8..15; one shfl_xor(16)
// completes the total (uniform across the wave).  EXEC must be all ones:
// callers invoke this unconditionally from full 256-thread blocks.
// ---------------------------------------------------------------------------
__device__ __forceinline__ float wave_sum64_wmma(float a0, float a1) {
  v2f a; a[0] = a0; a[1] = a1;
  v2f b; b[0] = 1.0f; b[1] = 1.0f;
  v8f c = {};
  c = __builtin_amdgcn_wmma_f32_16x16x4_f32(false, a, false, b, (short)0, c,
                                            false, false);
  float s = c[0] + c[1] + c[2] + c[3] + c[4] + c[5] + c[6] + c[7];
  s += __shfl_xor(s, 16, 32);
  return s;
}

__device__ __forceinline__ float sigmoidf_(float x) {
  return 1.0f / (1.0f + expf(-x));
}

// ---------------------------------------------------------------------------
// Zero the small stats/results region.
// ---------------------------------------------------------------------------
__global__ void k_zero(float* smallf, unsigned int* bsums) {
  int t = threadIdx.x;
  for (int i = t; i < 704; i += blockDim.x) smallf[i] = 0.0f;
  for (int i = t; i < NTILES; i += blockDim.x) bsums[i] = 0u;
}

// ---------------------------------------------------------------------------
// Pass 1: per-(b,id) {cnt, sum xyz, sum sigma} + per-b seed_bg.
// 2 voxels/thread so the seed_bg reduction maps onto the 16x4 WMMA tile.
// ---------------------------------------------------------------------------
__global__ void k_stats(const float* __restrict__ pred,
                        const int* __restrict__ inst,
                        const int* __restrict__ lab,
                        float* __restrict__ ACC, float* __restrict__ seed_bg) {
  const int b = blockIdx.y;
  const int t = threadIdx.x;
  const int j0 = blockIdx.x * (TPB * 2) + t;
  const int j1 = j0 + TPB;
  __shared__ float sacc[NID * 8];
  __shared__ float swsum[TPB / 32];
  for (int i = t; i < NID * 8; i += TPB) sacc[i] = 0.0f;
  __syncthreads();

  const float* pb = pred + (size_t)b * CH * VOL;
  const int* ib = inst + (size_t)b * VOL;
  const int* lb = lab + (size_t)b * VOL;
  float sq0 = 0.0f, sq1 = 0.0f;

  {
    int id = ib[j0];
    float sd = sigmoidf_(pb[(size_t)6 * VOL + j0]);
    if (lb[j0] == 0) sq0 = sd * sd;
    if (id >= 1 && id <= NID) {
      int base = (id - 1) * 8;
      int x = j0 % WW, y = (j0 / WW) % HH, z = j0 / (WW * HH);
      atomicAdd(&sacc[base + 0], 1.0f);
      atomicAdd(&sacc[base + 1], (float)x * (1.0f / (WW - 1)));
      atomicAdd(&sacc[base + 2], (float)y * (1.0f / (HH - 1)));
      atomicAdd(&sacc[base + 3], (float)z * (1.0f / (DD - 1)));
      atomicAdd(&sacc[base + 4], pb[(size_t)3 * VOL + j0]);
      atomicAdd(&sacc[base + 5], pb[(size_t)4 * VOL + j0]);
      atomicAdd(&sacc[base + 6], pb[(size_t)5 * VOL + j0]);
    }
  }
  {
    int id = ib[j1];
    float sd = sigmoidf_(pb[(size_t)6 * VOL + j1]);
    if (lb[j1] == 0) sq1 = sd * sd;
    if (id >= 1 && id <= NID) {
      int base = (id - 1) * 8;
      int x = j1 % WW, y = (j1 / WW) % HH, z = j1 / (WW * HH);
      atomicAdd(&sacc[base + 0], 1.0f);
      atomicAdd(&sacc[base + 1], (float)x * (1.0f / (WW - 1)));
      atomicAdd(&sacc[base + 2], (float)y * (1.0f / (HH - 1)));
      atomicAdd(&sacc[base + 3], (float)z * (1.0f / (DD - 1)));
      atomicAdd(&sacc[base + 4], pb[(size_t)3 * VOL + j1]);
      atomicAdd(&sacc[base + 5], pb[(size_t)4 * VOL + j1]);
      atomicAdd(&sacc[base + 6], pb[(size_t)5 * VOL + j1]);
    }
  }

  // seed_bg via WMMA wave reduction (all threads reach here: EXEC all ones)
  float ws = wave_sum64_wmma(sq0, sq1);
  int wave = t >> 5, lane = t & 31;
  if (lane == 0) swsum[wave] = ws;
  __syncthreads();
  if (t == 0) {
    float s = 0.0f;
    for (int w = 0; w < TPB / 32; w++) s += swsum[w];
    atomicAdd(&seed_bg[b], s);
  }
  if (t < NID * 8) atomicAdd(&ACC[b * NID * 8 + t], sacc[t]);
}

// ---------------------------------------------------------------------------
// Pass 2: finalize per-pair stats: center, s_mean, s=exp(10*s_mean), present.
// FIN layout per pair (12 f32): [cx,cy,cz, sx,sy,sz, m0,m1,m2, present, c, pad]
// ---------------------------------------------------------------------------
__global__ void k_finalize(const float* __restrict__ ACC, float* __restrict__ FIN) {
  int p = threadIdx.x;
  if (p >= NPAIR) return;
  const float* a = ACC + p * 8;
  float cnt = a[0];
  float c = fmaxf(cnt, 1.0f);
  float* f = FIN + p * 12;
  f[0] = a[1] / c; f[1] = a[2] / c; f[2] = a[3] / c;
  float m0 = a[4] / c, m1 = a[5] / c, m2 = a[6] / c;
  f[3] = expf(m0 * 10.0f); f[4] = expf(m1 * 10.0f); f[5] = expf(m2 * 10.0f);
  f[6] = m0; f[7] = m1; f[8] = m2;
  f[9] = (cnt > 0.0f) ? 1.0f : 0.0f;
  f[10] = c;
  f[11] = 0.0f;
}

// ---------------------------------------------------------------------------
// Pass 3: per-voxel var + seed_fg numerators (masks are disjoint: one pass).
// ---------------------------------------------------------------------------
__global__ void k_pairterms(const float* __restrict__ pred,
                            const int* __restrict__ inst,
                            const float* __restrict__ FIN,
                            float* __restrict__ var_num,
                            float* __restrict__ seedfg) {
  const int b = blockIdx.y;
  const int t = threadIdx.x;
  const int j = blockIdx.x * TPB + t;
  __shared__ float lv[NID], ls[NID];
  if (t < NID) { lv[t] = 0.0f; ls[t] = 0.0f; }
  __syncthreads();

  int id = inst[(size_t)b * VOL + j];
  if (id >= 1 && id <= NID) {
    const float* pb = pred + (size_t)b * CH * VOL;
    const float* f = FIN + (size_t)(b * NID + id - 1) * 12;
    float s0 = pb[(size_t)3 * VOL + j];
    float s1 = pb[(size_t)4 * VOL + j];
    float s2 = pb[(size_t)5 * VOL + j];
    float d0 = s0 - f[6], d1 = s1 - f[7], d2c = s2 - f[8];
    float var = d0 * d0 + d1 * d1 + d2c * d2c;
    int x = j % WW, y = (j / WW) % HH, z = j / (WW * HH);
    float ex = tanhf(pb[j]) + (float)x * (1.0f / (WW - 1));
    float ey = tanhf(pb[(size_t)VOL + j]) + (float)y * (1.0f / (HH - 1));
    float ez = tanhf(pb[(size_t)2 * VOL + j]) + (float)z * (1.0f / (DD - 1));
    float dx = ex - f[0], dy = ey - f[1], dz = ez - f[2];
    float dd = dx * dx * f[3] + dy * dy * f[4] + dz * dz * f[5];
    float dist = expf(-dd);
    float sd = sigmoidf_(pb[(size_t)6 * VOL + j]);
    float df = sd - dist;
    atomicAdd(&lv[id - 1], var);
    atomicAdd(&ls[id - 1], df * df);
  }
  __syncthreads();
  if (t < NID) {
    atomicAdd(&var_num[b * NID + t], lv[t]);
    atomicAdd(&seedfg[b * NID + t], ls[t]);
  }
}

// ---------------------------------------------------------------------------
// Per-pair key generation: 64-bit item = (descending-ordered err bits << 32)|m
// ---------------------------------------------------------------------------
__global__ void k_genkeys(const float* __restrict__ pred,
                          const int* __restrict__ inst,
                          const float* __restrict__ FIN,
                          unsigned long long* __restrict__ keys, int p) {
  int j = blockIdx.x * TPB + threadIdx.x;
  int b = p / NID, id = (p % NID) + 1;
  const float* pb = pred + (size_t)b * CH * VOL;
  const float* f = FIN + (size_t)p * 12;
  int x = j % WW, y = (j / WW) % HH, z = j / (WW * HH);
  float ex = tanhf(pb[j]) + (float)x * (1.0f / (WW - 1));
  float ey = tanhf(pb[(size_t)VOL + j]) + (float)y * (1.0f / (HH - 1));
  float ez = tanhf(pb[(size_t)2 * VOL + j]) + (float)z * (1.0f / (DD - 1));
  float dx = ex - f[0], dy = ey - f[1], dz = ez - f[2];
  float dd = dx * dx * f[3] + dy * dy * f[4] + dz * dz * f[5];
  float dist = expf(-dd);
  int m = (inst[(size_t)b * VOL + j] == id) ? 1 : 0;
  float signs = 2.0f * (float)m - 1.0f;
  float err = 1.0f - (2.0f * dist - 1.0f) * signs;
  unsigned int u = __float_as_uint(err);
  unsigned int asc = (u & 0x80000000u) ? ~u : (u | 0x80000000u);
  unsigned int dsc = ~asc;  // ascending sort of dsc == descending err
  keys[j] = ((unsigned long long)dsc << 32) | (unsigned long long)(unsigned)m;
}

// ---------------------------------------------------------------------------
// Radix sort: 4-bit digits over bits [32..63] (8 passes). hist[bin][tile].
// ---------------------------------------------------------------------------
__global__ void k_hist(const unsigned long long* __restrict__ src,
                       unsigned int* __restrict__ hist, int shift) {
  __shared__ unsigned int h[16];
  int t = threadIdx.x, tile = blockIdx.x;
  if (t < 16) h[t] = 0u;
  __syncthreads();
  size_t base = (size_t)tile * TILE + (size_t)t * IPT;
  __builtin_prefetch(&src[base], 0, 3);  // warm tile for the scatter pass
#pragma unroll
  for (int i = 0; i < IPT; i++) {
    unsigned int d = (unsigned int)(src[base + i] >> shift) & 15u;
    atomicAdd(&h[d], 1u);
  }
  __syncthreads();
  if (t < 16) hist[t * NTILES + tile] = h[t];
}

__global__ void k_scanhist(unsigned int* __restrict__ hist) {
  __shared__ unsigned int tot[16];
  __shared__ unsigned int base[16];
  int t = threadIdx.x;
  if (t < 16) {
    unsigned int run = 0;
    for (int k = 0; k < NTILES; k++) {
      unsigned int v = hist[t * NTILES + k];
      hist[t * NTILES + k] = run;
      run += v;
    }
    tot[t] = run;
  }
  __syncthreads();
  if (t == 0) {
    unsigned int r = 0;
    for (int d = 0; d < 16; d++) { base[d] = r; r += tot[d]; }
  }
  __syncthreads();
  if (t < 16) {
    unsigned int bs = base[t];
    for (int k = 0; k < NTILES; k++) hist[t * NTILES + k] += bs;
  }
}

__global__ void k_scatter(const unsigned long long* __restrict__ src,
                          unsigned long long* __restrict__ dst,
                          const unsigned int* __restrict__ hist, int shift) {
  __shared__ unsigned int scnt[16 * TPB];
  __shared__ unsigned int gofs[16];
  int t = threadIdx.x, tile = blockIdx.x;
  size_t base = (size_t)tile * TILE + (size_t)t * IPT;
  unsigned long long it[IPT];
  unsigned int dg[IPT];
#pragma unroll
  for (int i = 0; i < IPT; i++) {
    it[i] = src[base + i];
    dg[i] = (unsigned int)(it[i] >> shift) & 15u;
  }
  if (t < 16) gofs[t] = hist[t * NTILES + tile];
#pragma unroll
  for (int bin = 0; bin < 16; bin++) {
    unsigned int c = 0;
#pragma unroll
    for (int i = 0; i < IPT; i++) c += (dg[i] == (unsigned)bin) ? 1u : 0u;
    scnt[bin * TPB + t] = c;
  }
  __syncthreads();
  if (t < 16) {  // exclusive scan across threads, per bin (stable block rank)
    unsigned int run = 0;
    for (int k = 0; k < TPB; k++) {
      unsigned int v = scnt[t * TPB + k];
      scnt[t * TPB + k] = run;
      run += v;
    }
  }
  __syncthreads();
#pragma unroll
  for (int i = 0; i < IPT; i++) {
    unsigned int own = 0;
#pragma unroll
    for (int jj = 0; jj < IPT; jj++)
      if (jj < i) own += (dg[jj] == dg[i]) ? 1u : 0u;
    unsigned int r = gofs[dg[i]] + scnt[dg[i] * TPB + t] + own;
    dst[r] = it[i];
  }
}

// ---------------------------------------------------------------------------
// Label prefix-sum block sums + exclusive scan.
// ---------------------------------------------------------------------------
__global__ void k_blocksum(const unsigned long long* __restrict__ keys,
                           unsigned int* __restrict__ bsums) {
  __shared__ unsigned int s[TPB];
  int t = threadIdx.x, tile = blockIdx.x;
  size_t base = (size_t)tile * TILE + (size_t)t * IPT;
  unsigned int c = 0;
#pragma unroll
  for (int i = 0; i < IPT; i++) c += (unsigned int)(keys[base + i] & 1ull);
  s[t] = c;
  __syncthreads();
  for (int off = TPB / 2; off > 0; off >>= 1) {
    if (t < off) s[t] += s[t + off];
    __syncthreads();
  }
  if (t == 0) bsums[tile] = s[0];
}

__global__ void k_scanbsums(unsigned int* __restrict__ bsums) {
  unsigned int run = 0;
  for (int i = 0; i < NTILES; i++) {
    unsigned int v = bsums[i];
    bsums[i] = run;
    run += v;
  }
}

// ---------------------------------------------------------------------------
// Lovász scan over the sorted array:
//   F_k = inclusive cumsum of labels; jac_k = 1 - (G-F_k)/(G + (k+1) - F_k)
//   loss += relu(e_k) * (jac_k - jac_{k-1}),  jac_{-1} = 0.
// Final block reduction uses the WMMA row-sum trick.
// ---------------------------------------------------------------------------
__global__ void k_loss(const unsigned long long* __restrict__ keys,
                       const unsigned int* __restrict__ bsums,
                       const float* __restrict__ ACC,
                       float* __restrict__ lov, int p) {
  __shared__ unsigned int sx[TPB];
  __shared__ float wsum[TPB / 32];
  int t = threadIdx.x, tile = blockIdx.x;
  size_t base = (size_t)tile * TILE + (size_t)t * IPT;
  unsigned long long it[IPT];
  unsigned int c = 0;
#pragma unroll
  for (int i = 0; i < IPT; i++) {
    it[i] = keys[base + i];
    c += (unsigned int)(it[i] & 1ull);
  }
  sx[t] = c;
  __syncthreads();
  if (t == 0) {
    unsigned int run = 0;
    for (int k = 0; k < TPB; k++) {
      unsigned int v = sx[k];
      sx[k] = run;
      run += v;
    }
  }
  __syncthreads();

  const float G = ACC[p * 8];  // gts == fg count (exact integer in f32)
  unsigned int F = bsums[tile] + sx[t];
  float h0 = 0.0f, h1 = 0.0f;
#pragma unroll
  for (int i = 0; i < IPT; i++) {
    unsigned int m = (unsigned int)(it[i] & 1ull);
    F += m;  // inclusive
    size_t k = base + (size_t)i;
    unsigned int dsc = (unsigned int)(it[i] >> 32);
    unsigned int asc = ~dsc;
    unsigned int u = (asc & 0x80000000u) ? (asc ^ 0x80000000u) : ~asc;
    float e = __uint_as_float(u);
    float re = fmaxf(e, 0.0f);
    float kf = (float)k;
    float Ff = (float)F;
    float jac = 1.0f - (G - Ff) / (G + (kf + 1.0f) - Ff);
    float jacp;
    if (k == 0) {
      jacp = 0.0f;
    } else {
      float Fm = Ff - (float)m;
      jacp = 1.0f - (G - Fm) / (G + kf - Fm);
    }
    float contrib = re * (jac - jacp);
    if (i < IPT / 2) h0 += contrib; else h1 += contrib;
  }

  float ws = wave_sum64_wmma(h0, h1);  // all threads reach: EXEC all ones
  int wave = t >> 5, lane = t & 31;
  if (lane == 0) wsum[wave] = ws;
  __syncthreads();
  if (t == 0) {
    float s = 0.0f;
    for (int w = 0; w < TPB / 32; w++) s += wsum[w];
    atomicAdd(&lov[p], s);
  }
}

// ---------------------------------------------------------------------------
// Combine: per batch inst/var/seed losses, mean over batch.
// ---------------------------------------------------------------------------
__global__ void k_final(const float* __restrict__ FIN,
                        const float* __restrict__ var_num,
                        const float* __restrict__ seedfg,
                        const float* __restrict__ lov,
                        const float* __restrict__ seed_bg,
                        float* __restrict__ out) {
  if (threadIdx.x != 0 || blockIdx.x != 0) return;
  float total = 0.0f;
  for (int b = 0; b < BATCH; b++) {
    float obj = 0.0f, linst = 0.0f, lvar = 0.0f, lsfg = 0.0f;
    for (int i = 0; i < NID; i++) {
      int p = b * NID + i;
      const float* f = FIN + p * 12;
      float present = f[9];
      float cc = f[10];
      obj += present;
      linst += lov[p] * present;
      lvar += (var_num[p] / cc) * present;
      lsfg += seedfg[p] * present;  // FG_W = 1
    }
    float denom = fmaxf(obj, 1.0f);
    float seed_loss = (seed_bg[b] + lsfg) / (float)VOL;
    total += 1.0f * (linst / denom) + 10.0f * (lvar / denom) + 1.0f * seed_loss;
  }
  out[0] = total / (float)BATCH;
}

// ---------------------------------------------------------------------------
extern "C" void kernel_launch(void* const* d_in, const int* in_sizes, int n_in,
                              void* d_out, int out_size, void* d_ws,
                              size_t ws_size, hipStream_t stream) {
  (void)in_sizes; (void)n_in; (void)out_size; (void)ws_size;
  const float* pred = (const float*)d_in[0];
  const int* inst = (const int*)d_in[1];
  const int* lab = (const int*)d_in[2];
  // d_in[3] (center_images) unused by the reference computation.
  float* out = (float*)d_out;

  unsigned long long* bufA = (unsigned long long*)d_ws;
  unsigned long long* bufB = bufA + VOL;
  unsigned int* hist = (unsigned int*)(bufB + VOL);   // 16*NTILES
  unsigned int* bsums = hist + 16 * NTILES;           // NTILES
  float* smallf = (float*)(bsums + NTILES);
  float* ACC = smallf;            // 240
  float* FIN = smallf + 240;      // 360
  float* var_num = smallf + 600;  // 30
  float* seedfg = smallf + 630;   // 30
  float* lov = smallf + 660;      // 30
  float* seed_bg = smallf + 690;  // 2

  hipLaunchKernelGGL(k_zero, dim3(1), dim3(256), 0, stream, smallf, bsums);
  hipLaunchKernelGGL(k_stats, dim3(VOL / (TPB * 2), BATCH), dim3(TPB), 0,
                     stream, pred, inst, lab, ACC, seed_bg);
  hipLaunchKernelGGL(k_finalize, dim3(1), dim3(32), 0, stream, ACC, FIN);
  hipLaunchKernelGGL(k_pairterms, dim3(VOL / TPB, BATCH), dim3(TPB), 0, stream,
                     pred, inst, FIN, var_num, seedfg);

  for (int p = 0; p < NPAIR; p++) {
    hipLaunchKernelGGL(k_genkeys, dim3(VOL / TPB), dim3(TPB), 0, stream, pred,
                       inst, FIN, bufA, p);
    unsigned long long* src = bufA;
    unsigned long long* dst = bufB;
    for (int pass = 0; pass < 8; pass++) {
      int shift = 32 + pass * 4;
      hipLaunchKernelGGL(k_hist, dim3(NTILES), dim3(TPB), 0, stream, src, hist,
                         shift);
      hipLaunchKernelGGL(k_scanhist, dim3(1), dim3(32), 0, stream, hist);
      hipLaunchKernelGGL(k_scatter, dim3(NTILES), dim3(TPB), 0, stream, src,
                         dst, hist, shift);
      unsigned long long* tmp = src; src = dst; dst = tmp;
    }
    // 8 passes -> sorted result is back in bufA (== src)
    hipLaunchKernelGGL(k_blocksum, dim3(NTILES), dim3(TPB), 0, stream, src,
                       bsums);
    hipLaunchKernelGGL(k_scanbsums, dim3(1), dim3(1), 0, stream, bsums);
    hipLaunchKernelGGL(k_loss, dim3(NTILES), dim3(TPB), 0, stream, src, bsums,
                       ACC, lov, p);
  }

  hipLaunchKernelGGL(k_final, dim3(1), dim3(1), 0, stream, FIN, var_num, seedfg,
                     lov, seed_bg, out);
}